// DenseByItems_73048803770966
// MI455X (gfx1250) — compile-verified
//
#include <hip/hip_runtime.h>

typedef __attribute__((ext_vector_type(2))) float v2f;
typedef __attribute__((ext_vector_type(4))) float v4f;
typedef __attribute__((ext_vector_type(8))) float v8f;

#define BATCH_N 4096
#define EMB_D   128
#define K_ITEMS 200
#define N_TILES 13   // ceil(200/16)

__global__ __launch_bounds__(128)
void DenseByItems_wmma_kernel(const float* __restrict__ seq,   // [BATCH_N, EMB_D]
                              const int*   __restrict__ items, // [BATCH_N, K_ITEMS]
                              const float* __restrict__ V,     // [NUM_ITEMS, EMB_D]
                              const float* __restrict__ Bb,    // [NUM_ITEMS, 1]
                              float*       __restrict__ out)   // [BATCH_N, K_ITEMS]
{
    __shared__ float lds_seq[EMB_D];

    const int b   = blockIdx.x;
    const int tid = threadIdx.x;

    // Stage this batch row's query vector (512 B) into LDS once.
    lds_seq[tid] = seq[b * EMB_D + tid];
    __syncthreads();

    const int lane = tid & 31;
    const int wave = tid >> 5;
    const int half = lane >> 4;   // 0: lanes 0-15, 1: lanes 16-31
    const int nl   = lane & 15;   // item column within a 16-item tile

    // K-permuted operand staging: quad j for lane-half h covers K-words
    // [8j+4h .. 8j+4h+3].  WMMA (j,s) then consumes the even-aligned float2
    // slice {2s, 2s+1} of the quad in BOTH A and B, so the K-permutation is
    // consistent and the union over (j,s) covers K=0..127 exactly once.
    // A (query broadcast, 16x4 f32 layout) is tile-invariant -> hoist all 16
    // quads into registers via ds_load_b128 (same-address broadcast per half).
    v4f a[16];
    const v4f* lsp = (const v4f*)lds_seq;
    #pragma unroll
    for (int j = 0; j < 16; ++j) a[j] = lsp[2 * j + half];

    for (int t = wave; t < N_TILES; t += 4) {
        const int kk  = t * 16;
        const int kcl = min(kk + nl, K_ITEMS - 1);        // clamp tail
        const int idx = items[b * K_ITEMS + kcl];          // lanes l and l+16 share idx
        const v4f* vp = (const v4f*)(V + (long)idx * EMB_D);

        // Gather the whole 16-item tile as one clause of 16 b128 loads:
        // lane l (half h) reads V[item_{l&15}][8j+4h .. +3], 16 B per request.
        v4f bq[16];
        #pragma unroll
        for (int j = 0; j < 16; ++j) bq[j] = vp[2 * j + half];

        // Warm the next tile's rows while this tile's WMMAs run:
        // 2 prefetches/lane cover the 512 B row (lanes l, l+16 split the 4 lines).
        const int tn = t + 4;
        if (tn < N_TILES) {
            const int kcl2 = min(tn * 16 + nl, K_ITEMS - 1);
            const int idx2 = items[b * K_ITEMS + kcl2];
            const char* p  = (const char*)(V + (long)idx2 * EMB_D) + half * 256;
            __builtin_prefetch(p,       0, 3);
            __builtin_prefetch(p + 128, 0, 3);
        }

        v8f acc = {};
        #pragma unroll
        for (int j = 0; j < 16; ++j) {
            v2f a0 = { a[j].x,  a[j].y  }, a1 = { a[j].z,  a[j].w  };
            v2f b0 = { bq[j].x, bq[j].y }, b1 = { bq[j].z, bq[j].w };
            acc = __builtin_amdgcn_wmma_f32_16x16x4_f32(
                false, a0, false, b0, (short)0, acc, false, false);
            acc = __builtin_amdgcn_wmma_f32_16x16x4_f32(
                false, a1, false, b1, (short)0, acc, false, false);
        }

        // Every D row equals the 16 item scores; row M=0 is acc[0] in lanes 0-15.
        if (lane < 16 && (kk + nl) < K_ITEMS) {
            out[b * K_ITEMS + kk + nl] = acc[0] + Bb[idx];
        }
    }
}

extern "C" void kernel_launch(void* const* d_in, const int* in_sizes, int n_in,
                              void* d_out, int out_size, void* d_ws, size_t ws_size,
                              hipStream_t stream) {
    const float* seq   = (const float*)d_in[0];  // [4096, 128]
    const int*   items = (const int*)  d_in[1];  // [4096, 200]
    const float* V     = (const float*)d_in[2];  // [100000, 128]
    const float* Bb    = (const float*)d_in[3];  // [100000, 1]
    float*       out   = (float*)d_out;          // [4096, 200, 1]

    dim3 grid(BATCH_N);
    dim3 block(128);  // 4 waves of 32
    DenseByItems_wmma_kernel<<<grid, block, 0, stream>>>(seq, items, V, Bb, out);
}